// lossFunc_37735582663330
// MI455X (gfx1250) — compile-verified
//
#include <hip/hip_runtime.h>
#include <stdint.h>

// ---------------------------------------------------------------------------
// MI455X (gfx1250) implementation of the DKT BCE loss.
//
// Roofline: dominated by one streaming pass over `batch`
// ([2048,512,248] f32 = 1.04 GB). Rows of `batch` are one-hot, so the
// reference einsum collapses to a 4B gather from `pred` per (b,t)
// (~134 MB of touched cache lines) instead of a full 520 MB read.
// Total ~1.2 GB => ~50 us at 23.3 TB/s. Arithmetic is negligible, so
// WMMA is NOT profitable (it would force the full pred read and use only
// the diagonal of each 16x16 tile). The CDNA5 feature that pays off is
// the async global->LDS copy engine (ASYNCcnt): a deep, VGPR-free load
// pipeline feeding the one-hot scan. The stream is single-use, so it is
// tagged non-temporal (th:TH_LOAD_NT) to keep the 192 MB L2 clean.
// ---------------------------------------------------------------------------

#define NUM_Q    124
#define TWO_Q    248
#define MAX_T    512
#define NSTU     2048
#define TM1      (MAX_T - 1)                    // 511
#define FLOATS_PER_STU (MAX_T * TWO_Q)          // 126976
#define F4_PER_STU     (FLOATS_PER_STU / 4)     // 31744
#define THREADS  256
#define CHUNKS   (F4_PER_STU / THREADS)         // 124 chunks of 4 KB
#define DEPTH    8                              // async pipeline depth per wave

// gfx1250 async copy: 16B per lane, global -> LDS, tracked by ASYNCcnt.
// Non-temporal: batch is a read-once 1 GB stream.
__device__ __forceinline__ void async_g2l_b128(uint32_t lds_byte, uint64_t gaddr) {
    asm volatile("global_load_async_to_lds_b128 %0, %1, off th:TH_LOAD_NT"
                 :: "v"(lds_byte), "v"(gaddr) : "memory");
}
#define S_WAIT_ASYNCCNT(N) asm volatile("s_wait_asynccnt " #N ::: "memory")
#define S_WAIT_DSCNT0()    asm volatile("s_wait_dscnt 0" ::: "memory")

__device__ __forceinline__ void scan_one(float x, int ff, int* scol) {
    if (x >= 1.0f) {                 // batch entries are exactly 0.0 or 1.0
        int step = ff / TWO_Q;       // compiler lowers to mul-shift
        scol[step] = ff - step * TWO_Q;
    }
}

__device__ __forceinline__ float wave_sum(float v) {
    #pragma unroll
    for (int o = 16; o > 0; o >>= 1) v += __shfl_down(v, o, 32);
    return v;
}
__device__ __forceinline__ int wave_max(int v) {
    #pragma unroll
    for (int o = 16; o > 0; o >>= 1) { int t = __shfl_down(v, o, 32); v = v > t ? v : t; }
    return v;
}

__global__ __launch_bounds__(THREADS) void dkt_scan_kernel(
    const float* __restrict__ pred,   // [B, 512, 124]
    const float* __restrict__ batch,  // [B, 512, 248]
    float* __restrict__ out_p,        // [B, 511]
    float* __restrict__ out_a,        // [B, 511]
    float* __restrict__ ws_mean)      // [B] per-student mean BCE
{
    __shared__ float4 stage[8][DEPTH][32];   // 32 KB async staging
    __shared__ int    scol[MAX_T];           // one-hot column per step, -1 = padded
    __shared__ float  rsum[8];
    __shared__ int    rlast[8];

    const int tid  = threadIdx.x;
    const int w    = tid >> 5;
    const int lane = tid & 31;
    const int b    = blockIdx.x;

    for (int s = tid; s < MAX_T; s += THREADS) scol[s] = -1;
    __syncthreads();

    // ---- Phase 1: async-stream batch[b] through LDS, scan for one-hot hits.
    // chunk i covers float4 indices [i*256, i*256+256); this thread owns i*256+tid.
    const uint64_t gbase = (uint64_t)(uintptr_t)batch
                         + (uint64_t)b * (uint64_t)FLOATS_PER_STU * 4ull
                         + (uint64_t)tid * 16ull;
    const uint32_t lds0  = (uint32_t)(uintptr_t)&stage[w][0][lane];

    #pragma unroll
    for (int i = 0; i < DEPTH; ++i)
        async_g2l_b128(lds0 + (uint32_t)i * 512u, gbase + (uint64_t)i * 4096ull);

    int i = 0;
    for (; i < CHUNKS - DEPTH; ++i) {
        S_WAIT_ASYNCCNT(7);                       // chunk i has landed
        float4 v = stage[w][i & (DEPTH - 1)][lane];
        const int flat = (i * THREADS + tid) * 4;
        scan_one(v.x, flat + 0, scol);
        scan_one(v.y, flat + 1, scol);
        scan_one(v.z, flat + 2, scol);
        scan_one(v.w, flat + 3, scol);
        S_WAIT_DSCNT0();                          // slot reads done before reuse
        async_g2l_b128(lds0 + (uint32_t)(i & (DEPTH - 1)) * 512u,
                       gbase + (uint64_t)(i + DEPTH) * 4096ull);
    }
    S_WAIT_ASYNCCNT(0);
    for (; i < CHUNKS; ++i) {
        float4 v = stage[w][i & (DEPTH - 1)][lane];
        const int flat = (i * THREADS + tid) * 4;
        scan_one(v.x, flat + 0, scol);
        scan_one(v.y, flat + 1, scol);
        scan_one(v.z, flat + 2, scol);
        scan_one(v.w, flat + 3, scol);
    }
    __syncthreads();

    // ---- Phase 2: gather pred, emit p/a, accumulate BCE + last valid index.
    float lsum  = 0.0f;
    int   llast = -1;
    for (int t = tid; t < TM1; t += THREADS) {
        const int c1 = scol[t + 1];
        float pv = 0.0f, av = 0.0f;
        if (c1 >= 0) {
            const int q = (c1 >= NUM_Q) ? (c1 - NUM_Q) : c1;
            pv = __builtin_nontemporal_load(
                     &pred[((size_t)b * MAX_T + (size_t)t) * NUM_Q + q]);
            av = (c1 < NUM_Q) ? 1.0f : 0.0f;
            const float lp  = fmaxf(logf(pv),        -100.0f);
            const float l1p = fmaxf(logf(1.0f - pv), -100.0f);
            lsum += -(av * lp + (1.0f - av) * l1p);
            llast = t;   // valid steps are a prefix; max-reduce below anyway
        }
        const size_t o = (size_t)b * TM1 + (size_t)t;
        __builtin_nontemporal_store(pv, &out_p[o]);
        __builtin_nontemporal_store(av, &out_a[o]);
    }

    lsum  = wave_sum(lsum);
    llast = wave_max(llast);
    if (lane == 0) { rsum[w] = lsum; rlast[w] = llast; }
    __syncthreads();
    if (tid == 0) {
        float s = 0.0f; int L = -1;
        #pragma unroll
        for (int k = 0; k < 8; ++k) { s += rsum[k]; L = L > rlast[k] ? L : rlast[k]; }
        // L<0 => no valid steps: reference yields 0 contribution.
        ws_mean[b] = (L >= 0) ? (s / (float)(L + 1)) : 0.0f;
    }
}

// Deterministic final reduction of the 2048 per-student means.
__global__ __launch_bounds__(256) void dkt_final_kernel(
    const float* __restrict__ ws_mean, float* __restrict__ out_loss)
{
    __shared__ float sh[256];
    float v = 0.0f;
    for (int s = threadIdx.x; s < NSTU; s += 256) v += ws_mean[s];
    sh[threadIdx.x] = v;
    __syncthreads();
    for (int st = 128; st > 0; st >>= 1) {
        if (threadIdx.x < st) sh[threadIdx.x] += sh[threadIdx.x + st];
        __syncthreads();
    }
    if (threadIdx.x == 0) out_loss[0] = sh[0];
}

extern "C" void kernel_launch(void* const* d_in, const int* in_sizes, int n_in,
                              void* d_out, int out_size, void* d_ws, size_t ws_size,
                              hipStream_t stream) {
    (void)in_sizes; (void)n_in; (void)out_size; (void)ws_size;
    const float* pred  = (const float*)d_in[0];   // [2048,512,124] f32
    const float* batch = (const float*)d_in[1];   // [2048,512,248] f32
    float* out   = (float*)d_out;                 // [1 + 2048*511 + 2048*511]
    float* out_p = out + 1;
    float* out_a = out + 1 + (size_t)NSTU * TM1;
    float* ws    = (float*)d_ws;                  // 2048 floats of scratch

    dkt_scan_kernel<<<dim3(NSTU), dim3(THREADS), 0, stream>>>(
        pred, batch, out_p, out_a, ws);
    dkt_final_kernel<<<dim3(1), dim3(256), 0, stream>>>(ws, out + 0);
}